// MHCritic_14671608283444
// MI455X (gfx1250) — compile-verified
//
#include <hip/hip_runtime.h>

#define N_NODES 20000
#define N_EDGES 320000
#define E_TOT   (N_EDGES + N_NODES)   // self loops appended
#define D_IN    32
#define DIM     128
#define NV      8
#define DV      16
#define D_OUT   8

typedef float v2f __attribute__((ext_vector_type(2)));
typedef float v8f __attribute__((ext_vector_type(8)));

// ---------------------------------------------------------------------------
// Dense GEMM + bias + optional ReLU using V_WMMA_F32_16X16X4_F32.
// One wave32 computes one 16x16 tile of C = act(A[M,K] @ W[K,N] + b).
// A-matrix lane layout (16x4 f32): lane = 16*(k>=2) + m, vgpr = k&1.
// B-matrix lane layout (4x16 f32): lane = 16*(k>=2) + n, vgpr = k&1.
// C/D (16x16 f32, 8 VGPRs): vgpr r -> row r + 8*(lane/16), col lane%16.
// ---------------------------------------------------------------------------
__global__ void gemm_bias_act(const float* __restrict__ A,
                              const float* __restrict__ W,
                              const float* __restrict__ bias,
                              float* __restrict__ C,
                              int M, int K, int N, int relu)
{
    const int lane = threadIdx.x & 31;
    const int wavesPerBlock = blockDim.x >> 5;
    const int wave = blockIdx.x * wavesPerBlock + (threadIdx.x >> 5);
    const int ntiles = (N + 15) >> 4;
    const int mtiles = (M + 15) >> 4;
    if (wave >= ntiles * mtiles) return;           // wave-uniform: EXEC stays full
    const int m0 = (wave / ntiles) << 4;
    const int n0 = (wave % ntiles) << 4;

    const int half = lane >> 4;                    // selects K pair {0,1} vs {2,3}
    const int l16  = lane & 15;
    int rowA = m0 + l16; if (rowA >= M) rowA = M - 1;
    const int  colB  = n0 + l16;
    const bool colOK = colB < N;
    const int  colBc = colOK ? colB : (N - 1);

    v8f acc = {};
    for (int k = 0; k < K; k += 4) {
        const int kk = k + 2 * half;
        v2f a, b;
        a.x = A[rowA * K + kk];
        a.y = A[rowA * K + kk + 1];
        const float w0 = W[kk * N + colBc];
        const float w1 = W[(kk + 1) * N + colBc];
        b.x = colOK ? w0 : 0.0f;                   // cndmask select, no divergence
        b.y = colOK ? w1 : 0.0f;
        acc = __builtin_amdgcn_wmma_f32_16x16x4_f32(
                  false, a, false, b, (short)0, acc, false, false);
    }

    if (!colOK) return;
    const float bv = bias[colB];
#pragma unroll
    for (int r = 0; r < 8; ++r) {
        const int row = m0 + r + 8 * half;
        if (row < M) {
            float v = acc[r] + bv;
            if (relu) v = fmaxf(v, 0.0f);
            C[row * N + colB] = v;
        }
    }
}

// ---------------------------------------------------------------------------
__global__ void zero_f32(float* __restrict__ p, int n)
{
    int i = blockIdx.x * blockDim.x + threadIdx.x;
    if (i < n) p[i] = 0.0f;
}

__device__ __forceinline__ void edge_endpoints(const long long* __restrict__ ei,
                                               int e, int& s, int& d)
{
    if (e < N_EDGES) { s = (int)ei[e]; d = (int)ei[N_EDGES + e]; }
    else             { s = d = e - N_EDGES; }       // synthesized self loop
}

// Per (edge, head): logit = <Pt[dst,h,:], Ps[src,h,:]>, then segment max.
// ReLU inputs => logit >= 0 => float bits are monotone => int atomicMax exact,
// and zero-initialized max buffer is a correct identity.
__global__ void edge_logits(const long long* __restrict__ ei,
                            const float* __restrict__ Pt,
                            const float* __restrict__ Ps,
                            float* __restrict__ ework,
                            float* __restrict__ mmax)
{
    int idx = blockIdx.x * blockDim.x + threadIdx.x;
    if (idx >= E_TOT * NV) return;
    const int e = idx / NV, h = idx % NV;
    int s, d; edge_endpoints(ei, e, s, d);
    const float4* pt = (const float4*)(Pt + (size_t)d * DIM + h * DV);
    const float4* ps = (const float4*)(Ps + (size_t)s * DIM + h * DV);
    float acc = 0.0f;
#pragma unroll
    for (int i = 0; i < 4; ++i) {
        const float4 a = pt[i], b = ps[i];
        acc += a.x * b.x + a.y * b.y + a.z * b.z + a.w * b.w;
    }
    ework[idx] = acc;
    atomicMax((int*)(mmax + (size_t)d * NV + h), __float_as_int(acc));
}

// ex = exp(e - m[dst]); segment sum into s[dst].
__global__ void edge_expsum(const long long* __restrict__ ei,
                            float* __restrict__ ework,
                            const float* __restrict__ mmax,
                            float* __restrict__ ssum)
{
    int idx = blockIdx.x * blockDim.x + threadIdx.x;
    if (idx >= E_TOT * NV) return;
    const int e = idx / NV, h = idx % NV;
    int s, d; edge_endpoints(ei, e, s, d);
    const float ex = __expf(ework[idx] - mmax[(size_t)d * NV + h]);
    ework[idx] = ex;
    __hip_atomic_fetch_add(ssum + (size_t)d * NV + h, ex,
                           __ATOMIC_RELAXED, __HIP_MEMORY_SCOPE_AGENT);
}

// 16 lanes per edge, lane v: msg[v] = (1/NV) * sum_h Ph[src,h,v] * alpha[h];
// scatter-add into agg[dst, v]. Lane-consecutive v gives coalesced Ph reads.
__global__ void edge_aggregate(const long long* __restrict__ ei,
                               const float* __restrict__ Ph,
                               const float* __restrict__ ework,
                               const float* __restrict__ ssum,
                               float* __restrict__ agg)
{
    int gid = blockIdx.x * blockDim.x + threadIdx.x;
    const int e = gid >> 4;
    const int v = gid & 15;
    if (e >= E_TOT) return;
    int s, d; edge_endpoints(ei, e, s, d);
    const float* exe = ework + (size_t)e * NV;
    const float* sd  = ssum  + (size_t)d * NV;
    const float* ph  = Ph    + (size_t)s * DIM + v;
    float msg = 0.0f;
#pragma unroll
    for (int h = 0; h < NV; ++h) {
        const float alpha = exe[h] / (sd[h] + 1e-12f);
        msg += ph[h * DV] * alpha;
    }
    __hip_atomic_fetch_add(agg + (size_t)d * DV + v, msg * (1.0f / NV),
                           __ATOMIC_RELAXED, __HIP_MEMORY_SCOPE_AGENT);
}

// ---------------------------------------------------------------------------
extern "C" void kernel_launch(void* const* d_in, const int* in_sizes, int n_in,
                              void* d_out, int out_size, void* d_ws, size_t ws_size,
                              hipStream_t stream)
{
    const float*     x    = (const float*)d_in[0];
    const long long* ei   = (const long long*)d_in[1];
    const float*     W1   = (const float*)d_in[2];
    const float*     b1   = (const float*)d_in[3];
    const float*     W2   = (const float*)d_in[4];
    const float*     b2   = (const float*)d_in[5];
    const float*     Wout = (const float*)d_in[6];
    const float*     bout = (const float*)d_in[7];
    const float *Wt[2], *bt[2], *Wsr[2], *bsr[2], *Wh[2], *bh[2], *Wo[2], *bo[2];
    for (int l = 0; l < 2; ++l) {
        const int base = 8 + 8 * l;
        Wt[l]  = (const float*)d_in[base + 0];  bt[l]  = (const float*)d_in[base + 1];
        Wsr[l] = (const float*)d_in[base + 2];  bsr[l] = (const float*)d_in[base + 3];
        Wh[l]  = (const float*)d_in[base + 4];  bh[l]  = (const float*)d_in[base + 5];
        Wo[l]  = (const float*)d_in[base + 6];  bo[l]  = (const float*)d_in[base + 7];
    }

    // Workspace layout (floats). ~54.4 MB total.
    float* ws    = (float*)d_ws;
    float* h     = ws;                 // 20000*128
    float* Pt    = ws +  2560000;      // 20000*128 (also MLP temp)
    float* Ps    = ws +  5120000;      // 20000*128
    float* Ph    = ws +  7680000;      // 20000*128
    float* ework = ws + 10240000;      // 340000*8
    float* mmax  = ws + 12960000;      // 20000*8
    // ssum = ws + 13120000 (20000*8), agg = ws + 13280000 (20000*16): contiguous
    float* ssum  = ws + 13120000;
    float* agg   = ws + 13280000;

    auto gemm = [&](const float* A, const float* W, const float* b, float* C,
                    int M, int K, int N, int relu) {
        const int tiles  = ((M + 15) / 16) * ((N + 15) / 16);
        const int blocks = (tiles + 7) / 8;        // 8 waves / 256-thread block
        gemm_bias_act<<<blocks, 256, 0, stream>>>(A, W, b, C, M, K, N, relu);
    };

    // Embedding MLP
    gemm(x,  W1, b1, Pt, N_NODES, D_IN, DIM, 1);
    gemm(Pt, W2, b2, h,  N_NODES, DIM,  DIM, 1);

    const int ehThreads = E_TOT * NV;              // 2,720,000
    const int evThreads = E_TOT * 16;              // 5,440,000

    for (int l = 0; l < 2; ++l) {
        gemm(h, Wt[l],  bt[l],  Pt, N_NODES, DIM, DIM, 1);
        gemm(h, Wsr[l], bsr[l], Ps, N_NODES, DIM, DIM, 1);
        gemm(h, Wh[l],  bh[l],  Ph, N_NODES, DIM, DIM, 1);
        // zero mmax+ssum+agg (contiguous 640000 floats)
        zero_f32<<<(640000 + 255) / 256, 256, 0, stream>>>(mmax, 640000);
        edge_logits   <<<(ehThreads + 255) / 256, 256, 0, stream>>>(ei, Pt, Ps, ework, mmax);
        edge_expsum   <<<(ehThreads + 255) / 256, 256, 0, stream>>>(ei, ework, mmax, ssum);
        edge_aggregate<<<(evThreads + 255) / 256, 256, 0, stream>>>(ei, Ph, ework, ssum, agg);
        gemm(agg, Wo[l], bo[l], h, N_NODES, DV, DIM, 1);
    }

    // Output head (no relu), writes [20000, 8] directly to d_out
    gemm(h, Wout, bout, (float*)d_out, N_NODES, DIM, D_OUT, 0);
}